// HeteroGNNs_75316546502659
// MI455X (gfx1250) — compile-verified
//
#include <hip/hip_runtime.h>

typedef __attribute__((ext_vector_type(2))) float v2f;
typedef __attribute__((ext_vector_type(8))) float v8f;

#define NNODE 50000
#define FDIM  128
#define NEDGE 300000
#define NLAYER 2

// ---------------------------------------------------------------------------
// GEMM: H[N,128] = X[N,128] @ W[128,128] using V_WMMA_F32_16X16X4_F32.
// One wave (32 threads) computes a 16-row x 128-col output strip.
// A (16x4 f32): lane m=lane&15 holds rows; half=lane>>4 selects K pair.
//   a.x = X[r0+m][k0+2*half], a.y = X[r0+m][k0+2*half+1]
// B (4x16 f32): n=lane&15 cols; b.x = W[k0+2*half][c0+n], b.y = next K row.
// D (16x16 f32): acc[v] at row r0+v+8*half, col c0 + (lane&15).
// ---------------------------------------------------------------------------
__global__ __launch_bounds__(32)
void gemm_wmma_f32(const float* __restrict__ X, const float* __restrict__ W,
                   float* __restrict__ H) {
  const int r0   = blockIdx.x * 16;
  const int lane = threadIdx.x;
  const int half = lane >> 4;
  const int m    = lane & 15;

  v8f zero = {};
  v8f acc[8];
#pragma unroll
  for (int t = 0; t < 8; ++t) acc[t] = zero;

  const float* xrow = X + (size_t)(r0 + m) * FDIM;

  for (int k0 = 0; k0 < FDIM; k0 += 4) {
    const int ka = k0 + 2 * half;
    v2f a;
    a.x = xrow[ka];
    a.y = xrow[ka + 1];
#pragma unroll
    for (int t = 0; t < 8; ++t) {
      const int c0 = t * 16;
      v2f b;
      b.x = W[(size_t)ka * FDIM + c0 + m];
      b.y = W[(size_t)(ka + 1) * FDIM + c0 + m];
      acc[t] = __builtin_amdgcn_wmma_f32_16x16x4_f32(
          /*neg_a=*/false, a, /*neg_b=*/false, b,
          /*c_mod=*/(short)0, acc[t], /*reuse_a=*/false, /*reuse_b=*/false);
    }
  }

#pragma unroll
  for (int t = 0; t < 8; ++t) {
    const int c0 = t * 16;
#pragma unroll
    for (int v = 0; v < 8; ++v) {
      const int row = r0 + v + 8 * half;
      H[(size_t)row * FDIM + c0 + m] = acc[t][v];
    }
  }
}

// ---------------------------------------------------------------------------
// alpha[i] = dot(H[i,:], att)  -- one wave32 per node, float4 loads,
// wave32 butterfly reduction.
// ---------------------------------------------------------------------------
__global__ __launch_bounds__(128)
void alpha_dot(const float* __restrict__ H, const float* __restrict__ att,
               float* __restrict__ alpha, int n) {
  const int wave = threadIdx.x >> 5;
  const int lane = threadIdx.x & 31;
  const int node = blockIdx.x * 4 + wave;
  if (node >= n) return;
  const float4 h = ((const float4*)(H + (size_t)node * FDIM))[lane];
  const float4 a = ((const float4*)att)[lane];
  float s = h.x * a.x + h.y * a.y + h.z * a.z + h.w * a.w;
#pragma unroll
  for (int off = 16; off > 0; off >>= 1) s += __shfl_xor(s, off, 32);
  if (lane == 0) alpha[node] = s;
}

// monotonic float <-> uint mapping so atomicMax(uint) == float max
__device__ __forceinline__ unsigned enc_f(float f) {
  unsigned u = __float_as_uint(f);
  return (u & 0x80000000u) ? ~u : (u | 0x80000000u);
}
__device__ __forceinline__ float dec_f(unsigned k) {
  return __uint_as_float((k & 0x80000000u) ? (k & 0x7FFFFFFFu) : ~k);
}

__global__ void init_seg(unsigned* __restrict__ m, float* __restrict__ denom, int n) {
  int i = blockIdx.x * blockDim.x + threadIdx.x;
  if (i < n) { m[i] = 0u; denom[i] = 0.0f; }
}

// acc[i, f] = b1[f] + b2[f]  (pre-sum both convs' biases for aggr='sum')
__global__ void init_acc(float* __restrict__ acc, const float* __restrict__ b1,
                         const float* __restrict__ b2, int total) {
  int i = blockIdx.x * blockDim.x + threadIdx.x;
  if (i < total) acc[i] = b1[i & (FDIM - 1)] + b2[i & (FDIM - 1)];
}

// Pass 1: e = leaky_relu(as[src] + ad[dst]); segment max into m (encoded)
__global__ void edge_max(const float* __restrict__ as_, const float* __restrict__ ad_,
                         const int* __restrict__ src, const int* __restrict__ dst,
                         float* __restrict__ e, unsigned* __restrict__ m, int ne) {
  int k = blockIdx.x * blockDim.x + threadIdx.x;
  if (k >= ne) return;
  float x = as_[src[k]] + ad_[dst[k]];
  x = x > 0.0f ? x : 0.2f * x;   // leaky_relu, negative_slope = 0.2
  e[k] = x;
  atomicMax(&m[dst[k]], enc_f(x));
}

// Pass 2: p = exp(e - m[dst]) (in place over e); segment sum into denom
__global__ void edge_exp(float* __restrict__ e, const unsigned* __restrict__ m,
                         const int* __restrict__ dst, float* __restrict__ denom, int ne) {
  int k = blockIdx.x * blockDim.x + threadIdx.x;
  if (k >= ne) return;
  int d = dst[k];
  float p = expf(e[k] - dec_f(m[d]));
  e[k] = p;
  atomicAdd(&denom[d], p);
}

// Pass 3: acc[dst] += (p/denom[dst]) * h_src[src]; one wave32 per edge,
// float4 gather, 4 global f32 atomic adds per lane (128 feats / 32 lanes).
__global__ __launch_bounds__(256)
void edge_scatter(const float* __restrict__ p, const float* __restrict__ denom,
                  const int* __restrict__ src, const int* __restrict__ dst,
                  const float* __restrict__ h, float* __restrict__ acc, int ne) {
  const int wave = blockIdx.x * (blockDim.x >> 5) + (threadIdx.x >> 5);
  const int lane = threadIdx.x & 31;
  if (wave >= ne) return;
  const int s = src[wave];
  const int d = dst[wave];
  const float w = p[wave] / denom[d];
  const float4 hv = ((const float4*)(h + (size_t)s * FDIM))[lane];
  float* out = acc + (size_t)d * FDIM + lane * 4;
  atomicAdd(out + 0, w * hv.x);
  atomicAdd(out + 1, w * hv.y);
  atomicAdd(out + 2, w * hv.z);
  atomicAdd(out + 3, w * hv.w);
}

__global__ void relu_k(const float* __restrict__ in, float* __restrict__ out, int n) {
  int i = blockIdx.x * blockDim.x + threadIdx.x;
  if (i < n) out[i] = fmaxf(in[i], 0.0f);
}

// ---------------------------------------------------------------------------
// One GATConv: projections (WMMA), attention logits, 3-pass edge softmax +
// weighted scatter into `acc` (which was pre-initialized with the biases).
// ---------------------------------------------------------------------------
static void gat_conv_run(hipStream_t stream,
                         const float* x_src, const float* x_dst, int shared_w,
                         const float* W_src, const float* W_dst,
                         const float* att_s, const float* att_d,
                         const int* src, const int* dst,
                         float* h_src, float* h_dst,
                         float* alpha_s, float* alpha_d,
                         unsigned* m, float* denom, float* e_buf,
                         float* acc) {
  gemm_wmma_f32<<<NNODE / 16, 32, 0, stream>>>(x_src, W_src, h_src);
  if (shared_w) {
    h_dst = h_src;  // same x and W => same projection
  } else {
    gemm_wmma_f32<<<NNODE / 16, 32, 0, stream>>>(x_dst, W_dst, h_dst);
  }
  alpha_dot<<<NNODE / 4, 128, 0, stream>>>(h_src, att_s, alpha_s, NNODE);
  alpha_dot<<<NNODE / 4, 128, 0, stream>>>(h_dst, att_d, alpha_d, NNODE);
  init_seg<<<(NNODE + 255) / 256, 256, 0, stream>>>(m, denom, NNODE);
  edge_max<<<(NEDGE + 255) / 256, 256, 0, stream>>>(alpha_s, alpha_d, src, dst, e_buf, m, NEDGE);
  edge_exp<<<(NEDGE + 255) / 256, 256, 0, stream>>>(e_buf, m, dst, denom, NEDGE);
  edge_scatter<<<(NEDGE + 7) / 8, 256, 0, stream>>>(e_buf, denom, src, dst, h_src, acc, NEDGE);
}

extern "C" void kernel_launch(void* const* d_in, const int* in_sizes, int n_in,
                              void* d_out, int out_size, void* d_ws, size_t ws_size,
                              hipStream_t stream) {
  (void)in_sizes; (void)n_in; (void)out_size; (void)ws_size;

  const float* x_drug   = (const float*)d_in[0];
  const float* x_target = (const float*)d_in[1];
  const float* W_dd  = (const float*)d_in[2];
  const float* as_dd = (const float*)d_in[3];
  const float* ad_dd = (const float*)d_in[4];
  const float* b_dd  = (const float*)d_in[5];
  const float* Ws_dt = (const float*)d_in[6];
  const float* Wd_dt = (const float*)d_in[7];
  const float* as_dt = (const float*)d_in[8];
  const float* ad_dt = (const float*)d_in[9];
  const float* b_dt  = (const float*)d_in[10];
  const float* Ws_td = (const float*)d_in[11];
  const float* Wd_td = (const float*)d_in[12];
  const float* as_td = (const float*)d_in[13];
  const float* ad_td = (const float*)d_in[14];
  const float* b_td  = (const float*)d_in[15];
  const float* W_tt  = (const float*)d_in[16];
  const float* as_tt = (const float*)d_in[17];
  const float* ad_tt = (const float*)d_in[18];
  const float* b_tt  = (const float*)d_in[19];
  const int* ei_dd = (const int*)d_in[20];
  const int* ei_dt = (const int*)d_in[21];
  const int* ei_td = (const int*)d_in[22];
  const int* ei_tt = (const int*)d_in[23];

  // ---- workspace layout ----
  const size_t nf = (size_t)NNODE * FDIM;      // floats per node-feature matrix
  char* ws = (char*)d_ws;
  float* A_d    = (float*)(ws);                      // layer activations (drug)
  float* A_t    = A_d + nf;                          // layer activations (target)
  float* acc_d  = A_t + nf;                          // conv accumulator (drug)
  float* acc_t  = acc_d + nf;                        // conv accumulator (target)
  float* h0     = acc_t + nf;                        // projection buffer
  float* h1     = h0 + nf;                           // projection buffer
  float* alpha_s = h1 + nf;                          // [NNODE]
  float* alpha_d = alpha_s + NNODE;                  // [NNODE]
  unsigned* segm = (unsigned*)(alpha_d + NNODE);     // [NNODE]
  float* denom   = (float*)(segm + NNODE);           // [NNODE]
  float* e_buf   = denom + NNODE;                    // [NEDGE]

  const int NFTOT = NNODE * FDIM;

  for (int l = 0; l < NLAYER; ++l) {
    const float* xd = (l == 0) ? x_drug   : A_d;
    const float* xt = (l == 0) ? x_target : A_t;
    float* dest_d = (l == NLAYER - 1) ? (float*)d_out           : A_d;
    float* dest_t = (l == NLAYER - 1) ? (float*)d_out + NFTOT   : A_t;

    const size_t lW = (size_t)l * FDIM * FDIM;
    const size_t lH = (size_t)l * FDIM;

    // accumulators pre-loaded with summed biases of the two convs per dst set
    init_acc<<<(NFTOT + 255) / 256, 256, 0, stream>>>(acc_d, b_dd + lH, b_td + lH, NFTOT);
    init_acc<<<(NFTOT + 255) / 256, 256, 0, stream>>>(acc_t, b_dt + lH, b_tt + lH, NFTOT);

    // o_dd: (drug -> drug), shared W
    gat_conv_run(stream, xd, xd, /*shared_w=*/1, W_dd + lW, W_dd + lW,
                 as_dd + lH, ad_dd + lH, ei_dd, ei_dd + NEDGE,
                 h0, h1, alpha_s, alpha_d, segm, denom, e_buf, acc_d);
    // o_td: (target -> drug)
    gat_conv_run(stream, xt, xd, /*shared_w=*/0, Ws_td + lW, Wd_td + lW,
                 as_td + lH, ad_td + lH, ei_td, ei_td + NEDGE,
                 h0, h1, alpha_s, alpha_d, segm, denom, e_buf, acc_d);
    // o_dt: (drug -> target)
    gat_conv_run(stream, xd, xt, /*shared_w=*/0, Ws_dt + lW, Wd_dt + lW,
                 as_dt + lH, ad_dt + lH, ei_dt, ei_dt + NEDGE,
                 h0, h1, alpha_s, alpha_d, segm, denom, e_buf, acc_t);
    // o_tt: (target -> target), shared W
    gat_conv_run(stream, xt, xt, /*shared_w=*/1, W_tt + lW, W_tt + lW,
                 as_tt + lH, ad_tt + lH, ei_tt, ei_tt + NEDGE,
                 h0, h1, alpha_s, alpha_d, segm, denom, e_buf, acc_t);

    relu_k<<<(NFTOT + 255) / 256, 256, 0, stream>>>(acc_d, dest_d, NFTOT);
    relu_k<<<(NFTOT + 255) / 256, 256, 0, stream>>>(acc_t, dest_t, NFTOT);
  }
}